// kernel_generated_0_62904091017433
// MI455X (gfx1250) — compile-verified
//
#include <hip/hip_runtime.h>
#include <hip/hip_bf16.h>

typedef float v2f __attribute__((ext_vector_type(2)));
typedef float v8f __attribute__((ext_vector_type(8)));
typedef int   v4i_vs __attribute__((vector_size(16)));   // matches builtin param type

// Problem constants
#define BATCH   128
#define CCH     24     // channels in/out
#define HH      112    // l axis
#define WW      112    // m axis
#define GRP     2      // groups along l
#define GN      56     // rows per group
#define JF      14     // inner feature dim
#define TAPS    3
#define NB_ROWS 8      // output rows per block
#define HALO_R  10     // NB_ROWS + 2 halo
#define MSTRIDE 40     // weff M-stride: half-waves land 16 banks apart

// LDS layout: weff[3][24][40] then xs[10][24][112] (lr outer, ch, m inner)
#define WEFF_ELEMS (TAPS * CCH * MSTRIDE)         // 2880
#define XS_ELEMS   (HALO_R * CCH * WW)            // 26880

#if defined(__has_builtin)
#if __has_builtin(__builtin_amdgcn_global_load_async_to_lds_b128)
#define HAVE_ASYNC_LDS 1
#endif
#endif

__device__ __forceinline__ void wait_async_done() {
#if defined(HAVE_ASYNC_LDS)
#if __has_builtin(__builtin_amdgcn_s_wait_asynccnt)
    __builtin_amdgcn_s_wait_asynccnt(0);
#else
    asm volatile("s_wait_asynccnt 0x0" ::: "memory");
#endif
#endif
}

__global__ __launch_bounds__(256)
void fused_conv_wmma_kernel(const float* __restrict__ x,
                            const float* __restrict__ w0,
                            const float* __restrict__ w1,
                            float* __restrict__ out) {
    extern __shared__ float smem[];
    float* weff = smem;                 // [ (i*24+ch)*40 + c ]
    float* xs   = smem + WEFF_ELEMS;    // [ (lr*24+ch)*112 + m ]

    const int nb = blockIdx.x;          // 0..6  (row block)
    const int g  = blockIdx.y;          // 0..1  (group)
    const int kb = blockIdx.z;          // 0..127 (batch)
    const int tid = threadIdx.x;

    // ---- Stage x tile: 10 rows (1-row halo each side) x 24 ch x 112, b128 async ----
    // local row lr corresponds to group-row ng = nb*8 - 1 + lr; zero-pad outside [0,56)
    const int row0 = nb * NB_ROWS - 1;
    for (int t = tid; t < HALO_R * CCH * (WW / 4); t += 256) {
        const int m4 = (t % (WW / 4)) * 4;
        const int ch = (t / (WW / 4)) % CCH;
        const int lr = t / ((WW / 4) * CCH);
        const int ng = row0 + lr;
        float4* dst = (float4*)&xs[(lr * CCH + ch) * WW + m4];
        if (ng >= 0 && ng < GN) {
            const float* src =
                x + (((size_t)(kb * CCH + ch) * HH) + g * GN + ng) * WW + m4;
#if defined(HAVE_ASYNC_LDS)
            __builtin_amdgcn_global_load_async_to_lds_b128(
                (__attribute__((address_space(1))) v4i_vs*)src,
                (__attribute__((address_space(3))) v4i_vs*)dst, 0, 0);
#else
            *dst = *(const float4*)src;
#endif
        } else {
            float4 z = {0.0f, 0.0f, 0.0f, 0.0f};
            *dst = z;
        }
    }

    // ---- Build fused weights W_eff[i][c][ch] = sum_j w0[c,i,j]*w1[ch,j] ----
    // Stored as weff[(i*24+ch)*40 + c], c padded 24->40 with zeros (M padding).
    for (int t = tid; t < WEFF_ELEMS; t += 256) {
        const int c  = t % MSTRIDE;
        const int ch = (t / MSTRIDE) % CCH;
        const int i  = t / (MSTRIDE * CCH);
        float acc = 0.0f;
        if (c < CCH) {
            const float* w0p = w0 + (c * TAPS + i) * JF;
            const float* w1p = w1 + ch * JF;
            #pragma unroll
            for (int j = 0; j < JF; ++j) acc += w0p[j] * w1p[j];
        }
        weff[t] = acc;
    }

    wait_async_done();
    __syncthreads();

    // ---- Per-wave GEMM: D[32 x 112] = sum_i Weff_i[32x24] * Xrow_i[24x112] ----
    const int wv   = tid >> 5;        // wave id -> output row within block
    const int lane = tid & 31;
    const int half = lane >> 4;       // 0: lanes 0-15, 1: lanes 16-31
    const int lq   = lane & 15;

    const int n = nb * NB_ROWS + wv;  // group-row 0..55
    const int l = g * GN + n;         // global l

    for (int mt = 0; mt < 7; ++mt) {
        const int m0 = mt * 16;
        v8f acc0 = {};                // M-tile 0: c = 0..15
        v8f acc1 = {};                // M-tile 1: c = 16..31 (24..31 padded)

        #pragma unroll
        for (int i = 0; i < TAPS; ++i) {
            const int lr = wv + i;    // tap row: ng = n + i - 1 -> lr in [0,9]
            #pragma unroll
            for (int ks = 0; ks < 6; ++ks) {
                const int k0 = ks * 4;
                const int kk = k0 + (half << 1);   // this lane's K rows: kk, kk+1

                // B fragment (4x16 K x N): VGPR0 lanes0-15 K=k0 / lanes16-31 K=k0+2
                // half-waves are 224 words = 32 banks apart -> conflict-free
                v2f b;
                b.x = xs[(lr * CCH + kk) * WW + m0 + lq];
                b.y = xs[(lr * CCH + kk + 1) * WW + m0 + lq];

                // A fragments (16x4 M x K): lane lq = M; VGPR0 K=kk, VGPR1 K=kk+1
                // half-waves are 80 words = 16 banks apart -> conflict-free
                v2f a0, a1;
                a0.x = weff[(i * CCH + kk) * MSTRIDE + lq];
                a0.y = weff[(i * CCH + kk + 1) * MSTRIDE + lq];
                a1.x = weff[(i * CCH + kk) * MSTRIDE + 16 + lq];
                a1.y = weff[(i * CCH + kk + 1) * MSTRIDE + 16 + lq];

                acc0 = __builtin_amdgcn_wmma_f32_16x16x4_f32(
                    false, a0, false, b, (short)0, acc0, false, false);
                acc1 = __builtin_amdgcn_wmma_f32_16x16x4_f32(
                    false, a1, false, b, (short)0, acc1, false, false);
            }
        }

        // ---- Store with roll(+1) on m. C/D layout: VGPR r, lanes0-15 -> M=r,
        // lanes16-31 -> M=8+r, N = lane&15 + m0. Output is write-once -> NT stores.
        const int col  = m0 + lq;
        const int mout = (col + 1) % WW;
        #pragma unroll
        for (int r = 0; r < 8; ++r) {
            const int crow = r + (half << 3);
            // tile 0: c = crow (always < 16, valid)
            __builtin_nontemporal_store(
                acc0[r], &out[(((size_t)(kb * CCH + crow) * HH) + l) * WW + mout]);
            // tile 1: c = 16 + crow, valid only for c < 24
            const int c1 = 16 + crow;
            if (c1 < CCH) {
                __builtin_nontemporal_store(
                    acc1[r], &out[(((size_t)(kb * CCH + c1) * HH) + l) * WW + mout]);
            }
        }
    }
}

extern "C" void kernel_launch(void* const* d_in, const int* in_sizes, int n_in,
                              void* d_out, int out_size, void* d_ws, size_t ws_size,
                              hipStream_t stream) {
    const float* x  = (const float*)d_in[0];   // [128,24,112,112]
    const float* w0 = (const float*)d_in[1];   // [24,3,14]
    const float* w1 = (const float*)d_in[2];   // [24,14]
    float* out = (float*)d_out;                // [128,24,112,112]
    (void)in_sizes; (void)n_in; (void)out_size; (void)d_ws; (void)ws_size;

    const size_t lds_bytes = (size_t)(WEFF_ELEMS + XS_ELEMS) * sizeof(float);
    dim3 grid(GN / NB_ROWS, GRP, BATCH);   // (7, 2, 128)
    dim3 block(256);
    fused_conv_wmma_kernel<<<grid, block, lds_bytes, stream>>>(x, w0, w1, out);
}